// ImageEncoder_80960133530431
// MI455X (gfx1250) — compile-verified
//
#include <hip/hip_runtime.h>
#include <math.h>

typedef float v2f __attribute__((ext_vector_type(2)));
typedef float v8f __attribute__((ext_vector_type(8)));

#define B_ 64
#define S_ 1024
#define D_ 16
#define EPS_ 1e-5f

// ---------------------------------------------------------------------------
// Kernel 1: fused QKV projection + flash-style attention (L=2) + Wu proj +
// residual + LayerNorm.  One block per batch; K/V/Q tiles live in LDS.
// Writes h[B,S,16] into workspace.
// ---------------------------------------------------------------------------
__global__ __launch_bounds__(256) void attn_ln_kernel(
    const float* __restrict__ x,
    const float* __restrict__ Wq, const float* __restrict__ bq,
    const float* __restrict__ Wk, const float* __restrict__ bk,
    const float* __restrict__ Wv, const float* __restrict__ bv,
    const float* __restrict__ Wu, const float* __restrict__ bu,
    const float* __restrict__ ln_g, const float* __restrict__ ln_b,
    float* __restrict__ h_out)
{
    __shared__ float2 Qs[S_];
    __shared__ float2 Ks[S_];
    __shared__ float2 Vs[S_];

    const int b   = blockIdx.x;
    const int tid = threadIdx.x;
    const float* xb = x + (size_t)b * S_ * D_;

    // ---- stage 1: Q/K/V projections (D=16 -> L=2) ----
    for (int i = 0; i < S_ / 256; ++i) {
        const int tok = tid + 256 * i;
        const float* xr = xb + tok * D_;
        float q0 = bq[0], q1 = bq[1];
        float k0 = bk[0], k1 = bk[1];
        float v0 = bv[0], v1 = bv[1];
#pragma unroll
        for (int d = 0; d < D_; ++d) {
            const float xv = xr[d];
            q0 += xv * Wq[2 * d];  q1 += xv * Wq[2 * d + 1];
            k0 += xv * Wk[2 * d];  k1 += xv * Wk[2 * d + 1];
            v0 += xv * Wv[2 * d];  v1 += xv * Wv[2 * d + 1];
        }
        Qs[tok] = make_float2(q0, q1);
        Ks[tok] = make_float2(k0, k1);
        Vs[tok] = make_float2(v0, v1);
    }
    __syncthreads();

    // ---- stage 2: online-softmax attention + Wu + residual + LN ----
    for (int i = 0; i < S_ / 256; ++i) {
        const int s = tid + 256 * i;
        const float2 q = Qs[s];
        float m = -3.0e38f, l = 0.0f, a0 = 0.0f, a1 = 0.0f;
        for (int t = 0; t < S_; ++t) {
            const float2 k = Ks[t];
            const float2 v = Vs[t];
            const float sc = (q.x * k.x + q.y * k.y) * 0.25f; // 1/sqrt(16)
            const float mn = fmaxf(m, sc);
            const float corr = __expf(m - mn);
            const float p    = __expf(sc - mn);
            l  = l  * corr + p;
            a0 = a0 * corr + p * v.x;
            a1 = a1 * corr + p * v.y;
            m = mn;
        }
        const float inv_l = 1.0f / l;
        const float c0 = a0 * inv_l;
        const float c1 = a1 * inv_l;

        // ctx @ Wu + bu + x   (Wu is [2,16] row-major), then LayerNorm
        const float* xr = xb + s * D_;
        float u[D_];
        float mu = 0.0f;
#pragma unroll
        for (int d = 0; d < D_; ++d) {
            u[d] = c0 * Wu[d] + c1 * Wu[D_ + d] + bu[d] + xr[d];
            mu += u[d];
        }
        mu *= (1.0f / D_);
        float var = 0.0f;
#pragma unroll
        for (int d = 0; d < D_; ++d) {
            const float t = u[d] - mu;
            var += t * t;
        }
        var *= (1.0f / D_);
        const float rstd = rsqrtf(var + EPS_);
        float* ho = h_out + ((size_t)b * S_ + s) * D_;
#pragma unroll
        for (int d = 0; d < D_; ++d)
            ho[d] = (u[d] - mu) * rstd * ln_g[d] + ln_b[d];
    }
}

// ---------------------------------------------------------------------------
// Kernel 2: FFN chain with full-fp32 WMMA (V_WMMA_F32_16X16X4_F32).
// One wave (32 lanes) per 16-token tile:
//   d1 = h@W1+b1 ; d2 = d1@W2+b2 ; ff = d2@W3+b3 ;
//   e = LN(ff + h) ; out = e@Wo + bo
// Each 16x16x16 product = 4 chained K=4 WMMAs.  D-layout -> A-layout
// conversion goes through a padded LDS tile (single wave: DS is in-order).
// ---------------------------------------------------------------------------
#define TP 17   // padded LDS row stride (floats)

__global__ __launch_bounds__(32) void ffn_wmma_kernel(
    const float* __restrict__ h_in,
    const float* __restrict__ ln_g, const float* __restrict__ ln_b,
    const float* __restrict__ W1, const float* __restrict__ b1,
    const float* __restrict__ W2, const float* __restrict__ b2,
    const float* __restrict__ W3, const float* __restrict__ b3,
    const float* __restrict__ Wo, const float* __restrict__ bo,
    float* __restrict__ out)
{
    __shared__ float tA[16 * TP];
    __shared__ float tB[16 * TP];

    const int lane = threadIdx.x;
    const int n    = lane & 15;   // column / N index (also A-row index)
    const int half = lane >> 4;   // 0: lanes 0-15, 1: lanes 16-31
    const size_t base = (size_t)blockIdx.x * 16 * D_;  // 16 tokens

    // load h tile (row-major, padded) into LDS — coalesced
    {
        const float* hp = h_in + base;
#pragma unroll
        for (int i = 0; i < 8; ++i) {
            const int idx = lane + 32 * i;           // 0..255
            tA[(idx >> 4) * TP + (idx & 15)] = hp[idx];
        }
    }
    __syncthreads();

    // preload B operands: op g covers K rows 4g..4g+3.
    // VGPR0 = W[k][n], VGPR1 = W[k+1][n] with k = 4g + 2*half.
    v2f B1v[4], B2v[4], B3v[4], Bov[4];
#pragma unroll
    for (int g = 0; g < 4; ++g) {
        const int k = 4 * g + 2 * half;
        B1v[g].x = W1[k * 16 + n]; B1v[g].y = W1[(k + 1) * 16 + n];
        B2v[g].x = W2[k * 16 + n]; B2v[g].y = W2[(k + 1) * 16 + n];
        B3v[g].x = W3[k * 16 + n]; B3v[g].y = W3[(k + 1) * 16 + n];
        Bov[g].x = Wo[k * 16 + n]; Bov[g].y = Wo[(k + 1) * 16 + n];
    }

    // keep h in D-layout for the residual
    float hD[8];
#pragma unroll
    for (int r = 0; r < 8; ++r) hD[r] = tA[(r + 8 * half) * TP + n];

    // GEMM: A from LDS tile (row-major), B ops in regs, C init = bias[n]
    auto gemm = [&](const float* src, const v2f* bops, const float* bias) -> v8f {
        v8f c;
        const float bn = bias[n];
#pragma unroll
        for (int i = 0; i < 8; ++i) c[i] = bn;
#pragma unroll
        for (int g = 0; g < 4; ++g) {
            const int k = 4 * g + 2 * half;
            v2f a;
            a.x = src[n * TP + k];
            a.y = src[n * TP + k + 1];
            c = __builtin_amdgcn_wmma_f32_16x16x4_f32(
                    /*neg_a=*/false, a, /*neg_b=*/false, bops[g],
                    /*c_mod=*/(short)0, c, /*reuse_a=*/false, /*reuse_b=*/false);
        }
        return c;
    };
    // D-layout -> row-major LDS tile
    auto storeTile = [&](const v8f& d, float* dst) {
#pragma unroll
        for (int r = 0; r < 8; ++r) dst[(r + 8 * half) * TP + n] = d[r];
    };

    v8f d1 = gemm(tA, B1v, b1);
    storeTile(d1, tB);
    __syncthreads();
    v8f d2 = gemm(tB, B2v, b2);
    storeTile(d2, tA);
    __syncthreads();
    v8f ff = gemm(tA, B3v, b3);

    // residual + LayerNorm across the 16 feature lanes (xor-shuffle reduce)
    const float g_n = ln_g[n];
    const float bb_n = ln_b[n];
    v8f e;
#pragma unroll
    for (int r = 0; r < 8; ++r) {
        float v = ff[r] + hD[r];
        float s = v, q = v * v;
#pragma unroll
        for (int mask = 1; mask <= 8; mask <<= 1) {
            s += __shfl_xor(s, mask, 32);
            q += __shfl_xor(q, mask, 32);
        }
        const float mu  = s * (1.0f / 16.0f);
        const float var = q * (1.0f / 16.0f) - mu * mu;
        e[r] = (v - mu) * rsqrtf(var + EPS_) * g_n + bb_n;
    }
    storeTile(e, tB);
    __syncthreads();

    v8f o = gemm(tB, Bov, bo);

    float* op = out + base;
#pragma unroll
    for (int r = 0; r < 8; ++r) op[(r + 8 * half) * 16 + n] = o[r];
}

// ---------------------------------------------------------------------------
extern "C" void kernel_launch(void* const* d_in, const int* in_sizes, int n_in,
                              void* d_out, int out_size, void* d_ws, size_t ws_size,
                              hipStream_t stream) {
    const float* x    = (const float*)d_in[0];
    const float* Wq   = (const float*)d_in[1];
    const float* bq   = (const float*)d_in[2];
    const float* Wk   = (const float*)d_in[3];
    const float* bk   = (const float*)d_in[4];
    const float* Wv   = (const float*)d_in[5];
    const float* bv   = (const float*)d_in[6];
    const float* Wu   = (const float*)d_in[7];
    const float* bu   = (const float*)d_in[8];
    const float* ln_g = (const float*)d_in[9];
    const float* ln_b = (const float*)d_in[10];
    const float* W1   = (const float*)d_in[11];
    const float* b1   = (const float*)d_in[12];
    const float* W2   = (const float*)d_in[13];
    const float* b2   = (const float*)d_in[14];
    const float* W3   = (const float*)d_in[15];
    const float* b3   = (const float*)d_in[16];
    const float* Wo   = (const float*)d_in[17];
    const float* bo   = (const float*)d_in[18];

    float* h = (float*)d_ws;                 // [B,S,16] = 4 MB scratch
    float* out = (float*)d_out;

    attn_ln_kernel<<<B_, 256, 0, stream>>>(x, Wq, bq, Wk, bk, Wv, bv,
                                           Wu, bu, ln_g, ln_b, h);

    const int tiles = (B_ * S_) / 16;        // 4096 tiles of 16 tokens
    ffn_wmma_kernel<<<tiles, 32, 0, stream>>>(h, ln_g, ln_b,
                                              W1, b1, W2, b2, W3, b3,
                                              Wo, bo, out);
}